// HashedInterpolator_34926674051713
// MI455X (gfx1250) — compile-verified
//
#include <hip/hip_runtime.h>
#include <stdint.h>

// ---------------------------------------------------------------------------
// Hash-grid interpolation (Instant-NGP style) for MI455X / gfx1250.
//
// Memory-bound: 64MB hash table is L2-resident (192MB L2); streaming traffic
// (pos 24MB + out 32MB) bounds the kernel at ~2.4us @ 23.3 TB/s. Gathers are
// random 16B L2 hits. CDNA5-specific paths used:
//   * global_load_async_to_lds_b32 (GVS form: saddr + 32-bit voffset,
//     INST_OFFSET advances LDS and global sides together) + s_wait_asynccnt,
//     double-buffered position staging on the ASYNCcnt path, NT temporal hint
//   * non-temporal output stores (protect L2 residency of the table)
// ---------------------------------------------------------------------------

typedef float v4f __attribute__((ext_vector_type(4)));

#define HASH_MASK 0x3FFFFFu      // N_ENTRIES - 1, N_ENTRIES = 2^22
#define PRIME_Y   19349663u
#define PRIME_Z   83492791u
#define GRID_F    128.0f

#define BLOCK_THREADS 256
#define TILES         4
#define POINTS_PER_BLOCK (BLOCK_THREADS * TILES)

// Flat pointers into LDS carry the LDS byte offset in their low 32 bits
// (flat->LDS aperture mapping truncates to addr[31:0]).
__device__ __forceinline__ uint32_t lds_addr32(const void* p) {
    return (uint32_t)(uintptr_t)p;
}

// CDNA5 async global->LDS copies (per-lane b32), tracked by ASYNCcnt.
// GVS addressing: mem = SADDR + VADDR(u32) + IOFFSET, and IOFFSET is also
// added to the LDS destination, so one (lds, voff) pair stages 12 bytes.
// NT hint: positions are single-use streams; don't pollute L2.
__device__ __forceinline__ void async_stage_point(uint32_t lds, uint32_t voff,
                                                  uint64_t sbase) {
    asm volatile("global_load_async_to_lds_b32 %0, %1, %2 th:TH_LOAD_NT"
                 :: "v"(lds), "v"(voff), "s"(sbase) : "memory");
    asm volatile("global_load_async_to_lds_b32 %0, %1, %2 offset:4 th:TH_LOAD_NT"
                 :: "v"(lds), "v"(voff), "s"(sbase) : "memory");
    asm volatile("global_load_async_to_lds_b32 %0, %1, %2 offset:8 th:TH_LOAD_NT"
                 :: "v"(lds), "v"(voff), "s"(sbase) : "memory");
}

__device__ __forceinline__ void wait_asynccnt_le3() {
    asm volatile("s_wait_asynccnt 3" ::: "memory");
}
__device__ __forceinline__ void wait_asynccnt_0() {
    asm volatile("s_wait_asynccnt 0" ::: "memory");
}

// Per-point hash-grid interpolation (bit-faithful to the reference).
__device__ __forceinline__ v4f interp_point(float px, float py, float pz,
                                            const v4f* __restrict__ table) {
    // findGrid: p*128 is exact in fp32 (power-of-2 scale)
    const float gx = px * GRID_F, gy = py * GRID_F, gz = pz * GRID_F;
    const float flx = floorf(gx), fly = floorf(gy), flz = floorf(gz);
    const int ix = (int)flx, iy = (int)fly, iz = (int)flz;
    // == (p - lower/128) / (1/128) of the reference (scaling by 2^7 commutes
    // with rounding), selected by bit==0; bit==1 gets (1 - t).
    const float tx = gx - flx, ty = gy - fly, tz = gz - flz;

    // Hash: low 22 bits are exact in uint32 arithmetic.
    const uint32_t hx0 = (uint32_t)ix;
    const uint32_t hx1 = (uint32_t)(ix + 1);
    const uint32_t hy0 = (uint32_t)iy * PRIME_Y;
    const uint32_t hy1 = (uint32_t)(iy + 1) * PRIME_Y;
    const uint32_t hz0 = (uint32_t)iz * PRIME_Z;
    const uint32_t hz1 = (uint32_t)(iz + 1) * PRIME_Z;

    // Corner order = itertools.product([0,1], repeat=3): c = (bx, by, bz)
    const uint32_t i0 = (hx0 ^ hy0 ^ hz0) & HASH_MASK;
    const uint32_t i1 = (hx0 ^ hy0 ^ hz1) & HASH_MASK;
    const uint32_t i2 = (hx0 ^ hy1 ^ hz0) & HASH_MASK;
    const uint32_t i3 = (hx0 ^ hy1 ^ hz1) & HASH_MASK;
    const uint32_t i4 = (hx1 ^ hy0 ^ hz0) & HASH_MASK;
    const uint32_t i5 = (hx1 ^ hy0 ^ hz1) & HASH_MASK;
    const uint32_t i6 = (hx1 ^ hy1 ^ hz0) & HASH_MASK;
    const uint32_t i7 = (hx1 ^ hy1 ^ hz1) & HASH_MASK;

    // 8 independent b128 gathers (L2-resident table), issued back-to-back.
    const v4f v0 = table[i0];
    const v4f v1 = table[i1];
    const v4f v2 = table[i2];
    const v4f v3 = table[i3];
    const v4f v4 = table[i4];
    const v4f v5 = table[i5];
    const v4f v6 = table[i6];
    const v4f v7 = table[i7];

    // Faithful to reference weight assignment: bit==0 -> t, bit==1 -> 1-t.
    const float wx0 = tx, wx1 = 1.0f - tx;
    const float wy0 = ty, wy1 = 1.0f - ty;
    const float wz0 = tz, wz1 = 1.0f - tz;

    const float wxy00 = wx0 * wy0;
    const float wxy01 = wx0 * wy1;
    const float wxy10 = wx1 * wy0;
    const float wxy11 = wx1 * wy1;

    v4f acc = v0 * (wxy00 * wz0);
    acc += v1 * (wxy00 * wz1);
    acc += v2 * (wxy01 * wz0);
    acc += v3 * (wxy01 * wz1);
    acc += v4 * (wxy10 * wz0);
    acc += v5 * (wxy10 * wz1);
    acc += v6 * (wxy11 * wz0);
    acc += v7 * (wxy11 * wz1);
    return acc;
}

__global__ void __launch_bounds__(BLOCK_THREADS)
hashgrid_main(const float* __restrict__ pos,
              const v4f*  __restrict__ table,
              v4f*        __restrict__ out) {
    __shared__ float smem[2][BLOCK_THREADS * 3];
    const int tid = threadIdx.x;
    const long long base = (long long)blockIdx.x * POINTS_PER_BLOCK;
    const uint64_t posbase = (uint64_t)(uintptr_t)pos;

    // Byte offset of this thread's point in tile 0 (fits 32 bits: B*12 < 2^32)
    const uint32_t voff0 = (uint32_t)(base + tid) * 12u;

    // Prologue: asynchronously stage tile 0 into LDS buffer 0.
    async_stage_point(lds_addr32(&smem[0][tid * 3]), voff0, posbase);

#pragma unroll
    for (int t = 0; t < TILES; ++t) {
        const int buf = t & 1;
        if (t + 1 < TILES) {
            // Kick off next tile's staging, then wait for current tile only
            // (3 async ops per tile per wave; in-order completion).
            async_stage_point(lds_addr32(&smem[buf ^ 1][tid * 3]),
                              voff0 + (uint32_t)(t + 1) * (BLOCK_THREADS * 12u),
                              posbase);
            wait_asynccnt_le3();
        } else {
            wait_asynccnt_0();
        }

        const float px = smem[buf][tid * 3 + 0];
        const float py = smem[buf][tid * 3 + 1];
        const float pz = smem[buf][tid * 3 + 2];

        const v4f acc = interp_point(px, py, pz, table);

        // NT store: don't let the streamed 32MB output evict the L2-resident
        // 64MB hash table.
        __builtin_nontemporal_store(acc,
            &out[base + (long long)t * BLOCK_THREADS + tid]);
    }
}

// Bounds-checked fallback/tail kernel (not used for B = 2^21, kept for
// deterministic correctness at any size).
__global__ void __launch_bounds__(256)
hashgrid_tail(const float* __restrict__ pos,
              const v4f*  __restrict__ table,
              v4f*        __restrict__ out,
              long long n0, long long n) {
    const long long b = n0 + (long long)blockIdx.x * blockDim.x + threadIdx.x;
    if (b >= n) return;
    const float px = pos[b * 3 + 0];
    const float py = pos[b * 3 + 1];
    const float pz = pos[b * 3 + 2];
    const v4f acc = interp_point(px, py, pz, table);
    __builtin_nontemporal_store(acc, &out[b]);
}

extern "C" void kernel_launch(void* const* d_in, const int* in_sizes, int n_in,
                              void* d_out, int out_size, void* d_ws, size_t ws_size,
                              hipStream_t stream) {
    const float* pos   = (const float*)d_in[0];   // (B, 3) float32
    const v4f*   table = (const v4f*)d_in[1];     // (2^22, 4) float32
    v4f*         out   = (v4f*)d_out;             // (B, 4) float32

    const long long n     = (long long)in_sizes[0] / 3;
    const long long nmain = (n / POINTS_PER_BLOCK) * POINTS_PER_BLOCK;

    if (nmain > 0) {
        dim3 grid((unsigned)(nmain / POINTS_PER_BLOCK));
        hipLaunchKernelGGL(hashgrid_main, grid, dim3(BLOCK_THREADS), 0, stream,
                           pos, table, out);
    }
    if (n > nmain) {
        const long long tail = n - nmain;
        dim3 grid((unsigned)((tail + 255) / 256));
        hipLaunchKernelGGL(hashgrid_tail, grid, dim3(256), 0, stream,
                           pos, table, out, nmain, n);
    }
}